// MeshConvNet_75952201663132
// MI455X (gfx1250) — compile-verified
//
#include <hip/hip_runtime.h>
#include <math.h>

// ---- problem constants (from reference) ----
#define NN   32768      // total nodes
#define BB   16         // graphs
#define NPG  2048       // nodes per graph
#define EE   262144     // edges
#define NEGS 0.2f       // leaky relu slope
#define EPSB 1e-5f
#define NT   4          // column tiles per wave in WMMA GEMM

typedef __attribute__((ext_vector_type(2))) float v2f;
typedef __attribute__((ext_vector_type(8))) float v8f;

// ---------------- utility ----------------
__global__ void k_zero_i32(int* p, int n) {
    int i = blockIdx.x * blockDim.x + threadIdx.x;
    if (i < n) p[i] = 0;
}

// ---------------- CSR build (dest-sorted adjacency) ----------------
__global__ void k_count(const int* __restrict__ col, int* __restrict__ cnt, int n) {
    int e = blockIdx.x * blockDim.x + threadIdx.x;
    if (e < n) atomicAdd(&cnt[col[e]], 1);
}

// single block, 1024 threads, scans NN=32768 counts -> exclusive offsets (+copy to pos)
__global__ void k_scan(const int* __restrict__ cnt, int* __restrict__ off, int* __restrict__ pos) {
    __shared__ int part[1024];
    int t = threadIdx.x;
    int base = t * 32;
    int local[32];
    int s = 0;
    for (int j = 0; j < 32; ++j) { local[j] = s; s += cnt[base + j]; }
    part[t] = s;
    __syncthreads();
    for (int d = 1; d < 1024; d <<= 1) {
        int v = (t >= d) ? part[t - d] : 0;
        __syncthreads();
        part[t] += v;
        __syncthreads();
    }
    int ob = (t == 0) ? 0 : part[t - 1];
    for (int j = 0; j < 32; ++j) {
        int o = ob + local[j];
        off[base + j] = o;
        pos[base + j] = o;
    }
    if (t == 1023) off[NN] = part[1023];
}

__global__ void k_fill(const int* __restrict__ row, const int* __restrict__ col,
                       int* __restrict__ pos, int* __restrict__ csr, int n) {
    int e = blockIdx.x * blockDim.x + threadIdx.x;
    if (e < n) {
        int c = col[e];
        int slot = atomicAdd(&pos[c], 1);
        csr[slot] = row[e];
    }
}

// sort each node's neighbor list -> deterministic accumulation order across calls
__global__ void k_sortlists(const int* __restrict__ off, int* __restrict__ csr) {
    int i = blockIdx.x * blockDim.x + threadIdx.x;
    if (i >= NN) return;
    int s = off[i], e = off[i + 1];
    for (int a = s + 1; a < e; ++a) {
        int v = csr[a];
        int b = a - 1;
        while (b >= s && csr[b] > v) { csr[b + 1] = csr[b]; --b; }
        csr[b + 1] = v;
    }
}

__global__ void k_dinv(const int* __restrict__ off, float* __restrict__ dinv) {
    int i = blockIdx.x * blockDim.x + threadIdx.x;
    if (i >= NN) return;
    int d = off[i + 1] - off[i];
    dinv[i] = (d > 0) ? (1.0f / sqrtf((float)d)) : 0.0f;
}

// ---------------- weight transpose + zero-pad: Wt[n*fi+k] = n<fo ? W[k*fo+n] : 0 ----
__global__ void k_transpose_pad(const float* __restrict__ W, float* __restrict__ Wt,
                                int fi, int fo, int fopad) {
    int idx = blockIdx.x * blockDim.x + threadIdx.x;
    if (idx >= fopad * fi) return;
    int n = idx / fi;
    int k = idx - n * fi;
    Wt[idx] = (n < fo) ? W[(size_t)k * fo + n] : 0.f;
}

// ---------------- WMMA fp32 GEMM: Out(rows x fo) = X(rows x fi) @ W(fi x fo) ----
// Wt is the transposed, zero-padded weight (fopad x fi). One wave computes a
// 16-row x 64-col strip (NT=4 tiles) via V_WMMA_F32_16X16X4_F32. No guards in the
// hot loop: all loads unconditional (padding supplies zeros); store is guarded.
__global__ void k_gemm_wmma(const float* __restrict__ X, const float* __restrict__ Wt,
                            const float* __restrict__ bias, float* __restrict__ Out,
                            int rows, int fi, int fo, int ctg, int dorelu) {
    int wid = blockIdx.x * (blockDim.x >> 5) + (threadIdx.x >> 5);
    int rtiles = rows >> 4;
    if (wid >= rtiles * ctg) return;  // wave-uniform exit
    int tr = wid / ctg;
    int tc0 = (wid - tr * ctg) * (NT * 16);
    int lane = threadIdx.x & 31;
    int half = lane >> 4;              // A/B frag: lane holds k = 2*half + reg
    int lm = lane & 15;
    const float* __restrict__ xrow = X + (size_t)(tr * 16 + lm) * fi + 2 * half;
    const float* __restrict__ bp = Wt + (size_t)(tc0 + lm) * fi + 2 * half;
    size_t bstride = (size_t)16 * fi;  // next column tile
    v8f acc[NT];
    v8f zer = {0.f, 0.f, 0.f, 0.f, 0.f, 0.f, 0.f, 0.f};
#pragma unroll
    for (int t = 0; t < NT; ++t) acc[t] = zer;
    for (int k0 = 0; k0 < fi; k0 += 4) {
        v2f a = *(const v2f*)(xrow + k0);
#pragma unroll
        for (int t = 0; t < NT; ++t) {
            v2f b = *(const v2f*)(bp + (size_t)t * bstride + k0);
            acc[t] = __builtin_amdgcn_wmma_f32_16x16x4_f32(false, a, false, b, (short)0,
                                                           acc[t], false, false);
        }
    }
#pragma unroll
    for (int t = 0; t < NT; ++t) {
        int bcol = tc0 + t * 16 + lm;
        if (bcol < fo) {
            float bb = bias ? bias[bcol] : 0.f;
#pragma unroll
            for (int r = 0; r < 8; ++r) {  // D: reg r -> row m = 8*half + r
                int orow = tr * 16 + half * 8 + r;
                float v = acc[t][r] + bb;
                if (dorelu) v = fmaxf(v, 0.f);
                Out[(size_t)orow * fo + bcol] = v;
            }
        }
    }
}

// ---------------- GCN aggregation: out[i] = dinv[i]*sum_j dinv[j]*h[j] + b ----
__global__ void k_gcn_agg(const float* __restrict__ h, const int* __restrict__ off,
                          const int* __restrict__ src, const float* __restrict__ dinv,
                          const float* __restrict__ bias, float* __restrict__ out, int fo) {
    int i = blockIdx.x, ch = threadIdx.x;
    int s0 = off[i], s1 = off[i + 1];
    float di = dinv[i];
    float acc = 0.f;
    for (int j = s0; j < s1; ++j) {
        int s = src[j];
        acc += h[(size_t)s * fo + ch] * dinv[s];
    }
    out[(size_t)i * fo + ch] = acc * di + bias[ch];
}

// ---------------- BatchNorm (two-stage deterministic) + ReLU ----------------
__global__ void k_bn_partial(const float* __restrict__ x, float* __restrict__ ps,
                             float* __restrict__ pq, int fo) {
    int ch = threadIdx.x;
    int r0 = blockIdx.x * (NN / 256);
    float s = 0.f, q = 0.f;
    for (int r = 0; r < NN / 256; ++r) {
        float v = x[(size_t)(r0 + r) * fo + ch];
        s += v;
        q += v * v;
    }
    ps[blockIdx.x * fo + ch] = s;
    pq[blockIdx.x * fo + ch] = q;
}

__global__ void k_bn_finalize(const float* __restrict__ ps, const float* __restrict__ pq,
                              const float* __restrict__ g, const float* __restrict__ be,
                              float* __restrict__ mul, float* __restrict__ add, int fo) {
    int ch = threadIdx.x;
    float s = 0.f, q = 0.f;
    for (int b = 0; b < 256; ++b) { s += ps[b * fo + ch]; q += pq[b * fo + ch]; }
    float mu = s / (float)NN;
    float var = q / (float)NN - mu * mu;
    float m = g[ch] / sqrtf(var + EPSB);
    mul[ch] = m;
    add[ch] = be[ch] - mu * m;
}

__global__ void k_bn_relu(const float* __restrict__ in, const float* __restrict__ mul,
                          const float* __restrict__ add, float* __restrict__ out, int fo) {
    int idx = blockIdx.x * blockDim.x + threadIdx.x;
    int ch = idx % fo;
    out[idx] = fmaxf(in[idx] * mul[ch] + add[ch], 0.f);
}

// ---------------- ASAP: segment max over incoming neighbors + self ----------------
__global__ void k_segmax(const float* __restrict__ x, const int* __restrict__ off,
                         const int* __restrict__ src, float* __restrict__ xq, int fo) {
    int i = blockIdx.x, ch = threadIdx.x;
    int s0 = off[i], s1 = off[i + 1];
    float m = x[(size_t)i * fo + ch];
    for (int j = s0; j < s1; ++j) m = fmaxf(m, x[(size_t)src[j] * fo + ch]);
    xq[(size_t)i * fo + ch] = m;
}

// a1[i] = dot(xq[i], attW[:fo]) + attb ; a2[i] = dot(x[i], attW[fo:])
__global__ void k_att_dots(const float* __restrict__ xq, const float* __restrict__ x,
                           const float* __restrict__ attW, const float* __restrict__ attb,
                           float* __restrict__ a1, float* __restrict__ a2, int fo) {
    __shared__ float r1[256], r2[256];
    int i = blockIdx.x, ch = threadIdx.x;
    r1[ch] = xq[(size_t)i * fo + ch] * attW[ch];
    r2[ch] = x[(size_t)i * fo + ch] * attW[fo + ch];
    __syncthreads();
    for (int s = fo >> 1; s > 0; s >>= 1) {
        if (ch < s) { r1[ch] += r1[ch + s]; r2[ch] += r2[ch + s]; }
        __syncthreads();
    }
    if (ch == 0) { a1[i] = r1[0] + attb[0]; a2[i] = r2[0]; }
}

__device__ __forceinline__ float leaky(float v) { return v >= 0.f ? v : NEGS * v; }

// per-node softmax over incoming edges (+self) and weighted aggregation -> xp
__global__ void k_softmax_agg(const float* __restrict__ x, const int* __restrict__ off,
                              const int* __restrict__ src, const float* __restrict__ a1,
                              const float* __restrict__ a2, float* __restrict__ xp, int fo) {
    int i = blockIdx.x, ch = threadIdx.x;
    int s0 = off[i], s1 = off[i + 1];
    float base = a1[i];
    float lself = leaky(base + a2[i]);
    float smax = lself;
    for (int j = s0; j < s1; ++j) smax = fmaxf(smax, leaky(base + a2[src[j]]));
    float ssum = expf(lself - smax);
    for (int j = s0; j < s1; ++j) ssum += expf(leaky(base + a2[src[j]]) - smax);
    float acc = expf(lself - smax) * x[(size_t)i * fo + ch];
    for (int j = s0; j < s1; ++j) {
        int s = src[j];
        acc += expf(leaky(base + a2[s]) - smax) * x[(size_t)s * fo + ch];
    }
    xp[(size_t)i * fo + ch] = acc / ssum;
}

// LEConv scalar projections of xp
__global__ void k_le_dots(const float* __restrict__ xp, const float* __restrict__ w1,
                          const float* __restrict__ b1, const float* __restrict__ w2,
                          const float* __restrict__ w3, const float* __restrict__ b3,
                          float* __restrict__ aLE, float* __restrict__ bLE,
                          float* __restrict__ cLE, int fo) {
    __shared__ float r1[256], r2[256], r3[256];
    int i = blockIdx.x, ch = threadIdx.x;
    float v = xp[(size_t)i * fo + ch];
    r1[ch] = v * w1[ch];
    r2[ch] = v * w2[ch];
    r3[ch] = v * w3[ch];
    __syncthreads();
    for (int s = fo >> 1; s > 0; s >>= 1) {
        if (ch < s) { r1[ch] += r1[ch + s]; r2[ch] += r2[ch + s]; r3[ch] += r3[ch + s]; }
        __syncthreads();
    }
    if (ch == 0) { aLE[i] = r1[0] + b1[0]; bLE[i] = r2[0]; cLE[i] = r3[0] + b3[0]; }
}

// fit[i] = sigmoid( sum_{j in in(i)+self}(a[i]-bb[j]) + c3[i] )
__global__ void k_fitness(const int* __restrict__ off, const int* __restrict__ src,
                          const float* __restrict__ aLE, const float* __restrict__ bLE,
                          const float* __restrict__ cLE, float* __restrict__ fit) {
    int i = blockIdx.x * blockDim.x + threadIdx.x;
    if (i >= NN) return;
    int s0 = off[i], s1 = off[i + 1];
    float ai = aLE[i];
    float acc = ai - bLE[i];
    for (int j = s0; j < s1; ++j) acc += ai - bLE[src[j]];
    float z = acc + cLE[i];
    fit[i] = 1.f / (1.f + expf(-z));
}

// exact per-graph k-th largest via radix select on float bits (fit in (0,1) -> uint order)
__global__ void k_topk_kth(const float* __restrict__ fit, float* __restrict__ kth, int k) {
    __shared__ unsigned vals[NPG];
    __shared__ int scnt;
    int g = blockIdx.x;
    const float* f = fit + g * NPG;
    for (int j = threadIdx.x; j < NPG; j += 256) vals[j] = __float_as_uint(f[j]);
    __syncthreads();
    unsigned prefix = 0u;
    for (int b = 30; b >= 0; --b) {
        unsigned cand = prefix | (1u << b);
        if (threadIdx.x == 0) scnt = 0;
        __syncthreads();
        int c = 0;
        for (int j = threadIdx.x; j < NPG; j += 256) c += (vals[j] >= cand) ? 1 : 0;
        atomicAdd(&scnt, c);
        __syncthreads();
        if (scnt >= k) prefix = cand;
        __syncthreads();
    }
    if (threadIdx.x == 0) kth[g] = __uint_as_float(prefix);
}

__global__ void k_gate(const float* __restrict__ xp, const float* __restrict__ fit,
                       const float* __restrict__ kth, float* __restrict__ xout, int fo) {
    int idx = blockIdx.x * blockDim.x + threadIdx.x;
    int i = idx / fo;
    int g = i >> 11;  // NPG = 2048
    float fv = fit[i];
    float m = (fv >= kth[g]) ? fv : 0.f;
    xout[idx] = xp[idx] * m;
}

__global__ void k_meanpool(const float* __restrict__ x, float* __restrict__ xg, int fo) {
    int g = blockIdx.x, ch = threadIdx.x;
    float acc = 0.f;
    for (int r = 0; r < NPG; ++r) acc += x[(size_t)(g * NPG + r) * fo + ch];
    xg[g * fo + ch] = acc * (1.0f / (float)NPG);
}

// ---------------- host-side orchestration ----------------
extern "C" void kernel_launch(void* const* d_in, const int* in_sizes, int n_in,
                              void* d_out, int out_size, void* d_ws, size_t ws_size,
                              hipStream_t stream) {
    (void)in_sizes; (void)n_in; (void)out_size; (void)ws_size;
    char* w = (char*)d_ws;
    size_t cur = 0;
    auto alloc = [&](size_t bytes) -> void* {
        void* p = w + cur;
        cur = (cur + bytes + 255) & ~(size_t)255;
        return p;
    };
    float* bufX = (float*)alloc((size_t)NN * 256 * 4);
    float* bufH = (float*)alloc((size_t)NN * 256 * 4);
    float* bufG = (float*)alloc((size_t)NN * 256 * 4);
    float* wt   = (float*)alloc(256 * 256 * 4);   // transposed+padded weights
    int*   iOff = (int*)alloc((NN + 1) * 4);
    int*   iPos = (int*)alloc(NN * 4);
    int*   iCnt = (int*)alloc(NN * 4);
    int*   iSrc = (int*)alloc(EE * 4);
    float* dinv = (float*)alloc(NN * 4);
    float* a1   = (float*)alloc(NN * 4);
    float* a2   = (float*)alloc(NN * 4);
    float* aLE  = (float*)alloc(NN * 4);
    float* bLE  = (float*)alloc(NN * 4);
    float* cLE  = (float*)alloc(NN * 4);
    float* fit  = (float*)alloc(NN * 4);
    float* psum = (float*)alloc(256 * 256 * 4);
    float* pqsum= (float*)alloc(256 * 256 * 4);
    float* bnM  = (float*)alloc(256 * 4);
    float* bnA  = (float*)alloc(256 * 4);
    float* kth  = (float*)alloc(BB * 4);
    float* xg   = (float*)alloc(BB * 256 * 4);
    float* h1   = (float*)alloc(BB * 256 * 4);

    const float* x0  = (const float*)d_in[0];
    const int*   ei  = (const int*)d_in[1];
    const int*   row = ei;
    const int*   col = ei + EE;
    // d_in[2] = batch (layout known: g = node >> 11)

    auto gemm = [&](const float* X, const float* Wm, const float* bias, float* Out,
                    int rows, int fi, int fo, int relu) {
        int ct = (fo + 15) >> 4;
        int ctp = (ct + NT - 1) / NT * NT;   // pad column tiles to multiple of NT
        int fopad = ctp * 16;
        k_transpose_pad<<<(fopad * fi + 255) / 256, 256, 0, stream>>>(Wm, wt, fi, fo, fopad);
        int ctg = ctp / NT;
        int waves = (rows >> 4) * ctg;
        int blocks = (waves + 3) / 4;
        k_gemm_wmma<<<blocks, 128, 0, stream>>>(X, wt, bias, Out, rows, fi, fo, ctg, relu);
    };

    // ---- CSR build (once; edges constant across layers) ----
    k_zero_i32<<<NN / 256, 256, 0, stream>>>(iCnt, NN);
    k_count<<<EE / 256, 256, 0, stream>>>(col, iCnt, EE);
    k_scan<<<1, 1024, 0, stream>>>(iCnt, iOff, iPos);
    k_fill<<<EE / 256, 256, 0, stream>>>(row, col, iPos, iSrc, EE);
    k_sortlists<<<NN / 256, 256, 0, stream>>>(iOff, iSrc);
    k_dinv<<<NN / 256, 256, 0, stream>>>(iOff, dinv);

    static const int FI[3] = {16, 64, 128};
    static const int FO[3] = {64, 128, 256};
    static const int KP[3] = {1536, 1024, 512};

    const float* xcur = x0;
    for (int L = 0; L < 3; ++L) {
        int base = 3 + 13 * L;
        const float* Wl   = (const float*)d_in[base + 0];
        const float* bl   = (const float*)d_in[base + 1];
        const float* gl   = (const float*)d_in[base + 2];
        const float* bel  = (const float*)d_in[base + 3];
        const float* plW  = (const float*)d_in[base + 4];
        const float* plb  = (const float*)d_in[base + 5];
        const float* attW = (const float*)d_in[base + 6];
        const float* attb = (const float*)d_in[base + 7];
        const float* le1W = (const float*)d_in[base + 8];
        const float* le1b = (const float*)d_in[base + 9];
        const float* le2W = (const float*)d_in[base + 10];
        const float* le3W = (const float*)d_in[base + 11];
        const float* le3b = (const float*)d_in[base + 12];
        int fi = FI[L], fo = FO[L];

        // GCN: h = x @ W ; aggregate with symmetric norm ; + bias
        gemm(xcur, Wl, nullptr, bufH, NN, fi, fo, 0);
        k_gcn_agg<<<NN, fo, 0, stream>>>(bufH, iOff, iSrc, dinv, bl, bufG, fo);
        // BN (train-mode batch stats) + ReLU -> bufX
        k_bn_partial<<<256, fo, 0, stream>>>(bufG, psum, pqsum, fo);
        k_bn_finalize<<<1, fo, 0, stream>>>(psum, pqsum, gl, bel, bnM, bnA, fo);
        k_bn_relu<<<(NN * fo) / 256, 256, 0, stream>>>(bufG, bnM, bnA, bufX, fo);
        // ASAP
        k_segmax<<<NN, fo, 0, stream>>>(bufX, iOff, iSrc, bufH, fo);
        gemm(bufH, plW, plb, bufG, NN, fo, fo, 0);  // xq
        k_att_dots<<<NN, fo, 0, stream>>>(bufG, bufX, attW, attb, a1, a2, fo);
        k_softmax_agg<<<NN, fo, 0, stream>>>(bufX, iOff, iSrc, a1, a2, bufH, fo);  // xp
        k_le_dots<<<NN, fo, 0, stream>>>(bufH, le1W, le1b, le2W, le3W, le3b, aLE, bLE, cLE, fo);
        k_fitness<<<NN / 256, 256, 0, stream>>>(iOff, iSrc, aLE, bLE, cLE, fit);
        k_topk_kth<<<BB, 256, 0, stream>>>(fit, kth, KP[L]);
        k_gate<<<(NN * fo) / 256, 256, 0, stream>>>(bufH, fit, kth, bufX, fo);
        xcur = bufX;
    }

    // readout
    k_meanpool<<<BB, 256, 0, stream>>>(bufX, xg, 256);
    const float* fc1W = (const float*)d_in[42];
    const float* fc1b = (const float*)d_in[43];
    const float* fc2W = (const float*)d_in[44];
    const float* fc2b = (const float*)d_in[45];
    gemm(xg, fc1W, fc1b, h1, BB, 256, 256, 1);
    gemm(h1, fc2W, fc2b, (float*)d_out, BB, 256, 40, 0);
}